// DPAAUser3D_26001732010123
// MI455X (gfx1250) — compile-verified
//
#include <hip/hip_runtime.h>

typedef __attribute__((ext_vector_type(2))) float v2f;
typedef __attribute__((ext_vector_type(8))) float v8f;

#define CH_STRIDE (128*128*128)   // per-channel voxel count
#define SLICE     (128*128)       // one D-slice
#define NVOX      (2*8*128*128*128)

// ---------------------------------------------------------------------------
// Kernel 0: zero the adj counters (2*8*16^3 = 65536 uints in d_ws)
// ---------------------------------------------------------------------------
__global__ void k_zero(unsigned int* __restrict__ ws) {
    int i = blockIdx.x * blockDim.x + threadIdx.x;
    if (i < 65536) ws[i] = 0u;
}

// ---------------------------------------------------------------------------
// Kernel 1: per-voxel channel argmax + 8x8x8 pooled histogram.
// grid.x = B * D * (H/8) = 2*128*16 = 4096, block = 1024 threads
// (one workgroup = 8 contiguous H-rows of one D-slice = 1024 contiguous floats
//  per channel -> fully coalesced; x gets pulled into the 192MB L2 here and
//  stays resident for kernel 2).
// ---------------------------------------------------------------------------
__global__ void k_argmax_pool(const float* __restrict__ x,
                              unsigned int* __restrict__ cnt) {
    __shared__ unsigned int hist[128];          // [pw 0..15][c 0..7]
    int tid = threadIdx.x;
    if (tid < 128) hist[tid] = 0u;
    __syncthreads();

    int wg = blockIdx.x;
    int ph = wg & 15;                           // H pool-block
    int d  = (wg >> 4) & 127;                   // voxel depth D
    int b  = wg >> 11;                          // batch
    int hrow = (ph << 3) + (tid >> 7);          // H
    int w    = tid & 127;                       // W
    long base = (long)b * 8 * CH_STRIDE + (long)d * SLICE + hrow * 128 + w;

    // argmax over 8 channels, first-index-wins on ties (matches jnp.argmax)
    float best = x[base];
    int idx = 0;
    #pragma unroll
    for (int c = 1; c < 8; ++c) {
        float v = x[base + (long)c * CH_STRIDE];
        if (v > best) { best = v; idx = c; }
    }
    atomicAdd(&hist[((w >> 3) << 3) + idx], 1u);
    __syncthreads();

    if (tid < 128) {
        int pw = tid >> 3, c = tid & 7;
        unsigned int v = hist[(pw << 3) + c];
        if (v) {
            // adj layout: (((b*8+c)*16 + pd)*16 + ph)*16 + pw, pd = D/8
            int gi = (((b * 8 + c) * 16 + (d >> 3)) * 16 + ph) * 16 + pw;
            atomicAdd(&cnt[gi], v);
        }
    }
}

// ---------------------------------------------------------------------------
// Kernel 2: WMMA rank-1 correction tiles + fused apply.
// grid.x = B*C*D = 2048, block = 256 (8 waves). Each wave does 8
// v_wmma_f32_16x16x4_f32 ops; each WMMA yields corr for 2 full H-rows
// (256 voxels). Accumulator layout: lane l, vgpr v holds
//   H = (h0 + l/16)*16 + kh,  W = v*16 + (l%16)
// -> per-VGPR half-wave spans are contiguous 64B segments (coalesced).
// ---------------------------------------------------------------------------
__global__ void k_apply(const float* __restrict__ x,
                        const float* __restrict__ att,
                        const unsigned int* __restrict__ cnt,
                        float* __restrict__ out) {
    __shared__ float att_s[64];    // att for this d-block: i = d*64 + (h*8+w)
    __shared__ float adj_s[256];   // adj for this kd:      p = kd*256 + kh*16+kw
    int tid = threadIdx.x;
    int wg  = blockIdx.x;
    int D   = wg & 127;
    int bc  = wg >> 7;             // b*8 + c
    int dblk = D >> 4;             // patch index along depth
    int kd   = D & 15;

    if (tid < 64) att_s[tid] = att[bc * 512 + dblk * 64 + tid];
    adj_s[tid] = (float)cnt[bc * 4096 + kd * 256 + tid] * (1.0f / 512.0f);
    __syncthreads();

    int l  = tid & 31;             // lane in wave32
    int lo = l & 15;
    int hi = l >> 4;
    int wave = tid >> 5;

    long slice = (long)wg * SLICE;
    const float* xs = x + slice;
    float* o0 = out + slice;                    // out0 = x*(1+corr)^2
    float* o1 = out + (long)NVOX + slice;       // out1 = corr

    #pragma unroll
    for (int j = 0; j < 8; ++j) {
        int t  = wave + (j << 3);               // tile id 0..63
        int kh = t & 15;
        int h0 = (t >> 4) << 1;                 // H-patch pair base (0,2,4,6)

        // A: K=0 column = att[i0 + M] (lanes 0-15, V0); all other K zero.
        // B: K=0 row    = adj[p0 + N] (lanes 0-15, V0); all other K zero.
        float av = att_s[h0 * 8 + lo];
        float bv = adj_s[kh * 16 + lo];
        v2f a;  a.x  = (l < 16) ? av : 0.0f;  a.y  = 0.0f;
        v2f bb; bb.x = (l < 16) ? bv : 0.0f;  bb.y = 0.0f;
        v8f c = {};
        // D[M,N] = att[i0+M] * adj[p0+N]
        c = __builtin_amdgcn_wmma_f32_16x16x4_f32(false, a, false, bb,
                                                  (short)0, c, false, false);

        int row  = ((h0 + hi) << 4) + kh;       // H
        int boff = row * 128 + lo;
        #pragma unroll
        for (int v = 0; v < 8; ++v) {
            int off = boff + (v << 4);          // W = v*16 + lo
            float xv   = xs[off];               // RT load: hits L2 (warmed by k1)
            float corr = c[v];
            float y    = fmaf(corr, xv, xv);    // x + corr*x
            float r    = fmaf(corr, y, y);      // y + corr*y
            __builtin_nontemporal_store(r,    o0 + off);  // NT: don't evict x
            __builtin_nontemporal_store(corr, o1 + off);
        }
    }
}

// ---------------------------------------------------------------------------
extern "C" void kernel_launch(void* const* d_in, const int* in_sizes, int n_in,
                              void* d_out, int out_size, void* d_ws, size_t ws_size,
                              hipStream_t stream) {
    const float* x   = (const float*)d_in[0];   // [2,8,128,128,128] f32
    const float* att = (const float*)d_in[1];   // [2,8,512,1] f32
    float* out = (float*)d_out;                 // out0 (33.5M) ++ corr (33.5M)
    unsigned int* cnt = (unsigned int*)d_ws;    // 65536 uints (256 KB)

    k_zero<<<256, 256, 0, stream>>>(cnt);
    k_argmax_pool<<<4096, 1024, 0, stream>>>(x, cnt);
    k_apply<<<2048, 256, 0, stream>>>(x, att, cnt, out);
}